// QIntSoftmax_22866405884465
// MI455X (gfx1250) — compile-verified
//
#include <hip/hip_runtime.h>
#include <hip/hip_bf16.h>

typedef __attribute__((ext_vector_type(4))) float v4f;

// Group-of-32 xor swizzle: offset = and_mask(0x1f) | or_mask(0)<<5 | xor_mask<<10
#define SWZ_XOR(v, m) __builtin_amdgcn_ds_swizzle((v), (0x1f | ((m) << 10)))

__device__ __forceinline__ float wave32_max(float v) {
    v = fmaxf(v, __int_as_float(SWZ_XOR(__float_as_int(v), 16)));
    v = fmaxf(v, __int_as_float(SWZ_XOR(__float_as_int(v), 8)));
    v = fmaxf(v, __int_as_float(SWZ_XOR(__float_as_int(v), 4)));
    v = fmaxf(v, __int_as_float(SWZ_XOR(__float_as_int(v), 2)));
    v = fmaxf(v, __int_as_float(SWZ_XOR(__float_as_int(v), 1)));
    return v;  // butterfly: every lane holds the row max
}

__device__ __forceinline__ float wave32_sum(float v) {
    v += __int_as_float(SWZ_XOR(__float_as_int(v), 16));
    v += __int_as_float(SWZ_XOR(__float_as_int(v), 8));
    v += __int_as_float(SWZ_XOR(__float_as_int(v), 4));
    v += __int_as_float(SWZ_XOR(__float_as_int(v), 2));
    v += __int_as_float(SWZ_XOR(__float_as_int(v), 1));
    return v;  // every lane holds the row sum
}

// One wave32 per 1024-element row; 8 rows per 256-thread block.
__global__ __launch_bounds__(256)
void qintsoftmax_kernel(const float* __restrict__ x,
                        const float* __restrict__ scale,
                        float* __restrict__ out, int rows) {
    const int wave = threadIdx.x >> 5;
    const int lane = threadIdx.x & 31;
    const int row  = blockIdx.x * 8 + wave;
    if (row >= rows) return;

    // Uniform constants derived from the quantization scale (s > 0).
    const float s      = scale[0];
    const float inv_s  = 1.0f / s;
    const float x0i    = floorf(-0.6931f / s);                   // floor(-ln2 / s), negative
    const float inv_x0 = 1.0f / x0i;
    const float bI     = floorf((0.96963238f / 0.35815147f) / s);
    const float cI     = floorf((1.0f / 0.35815147f) / (s * s));
    const float nx0    = 30.0f * x0i;                            // n * x0_int clamp

    const size_t base = (size_t)row * 1024 + (size_t)(lane * 4);
    const float* xp = x + base;
    float*       op = out + base;

    // ---- Single HBM read of the row into VGPRs (8 x global_load_b128, NT) ----
    v4f v[8];
#pragma unroll
    for (int i = 0; i < 8; ++i)
        v[i] = __builtin_nontemporal_load((const v4f*)(xp + i * 128));

    // ---- Pass 1: raw row max (1 v_max per element); scale once after reduce ----
    float m = -__builtin_huge_valf();
#pragma unroll
    for (int i = 0; i < 8; ++i) {
#pragma unroll
        for (int c = 0; c < 4; ++c)
            m = fmaxf(m, v[i][c]);
    }
    m = wave32_max(m) * inv_s;   // == max(x / s) since s > 0

    // ---- Pass 2: integer-poly exp approximation in place, row sum ----
    float sum = 0.0f;
#pragma unroll
    for (int i = 0; i < 8; ++i) {
#pragma unroll
        for (int c = 0; c < 4; ++c) {
            float xi = fmaf(v[i][c], inv_s, -m);       // x/s - max
            xi = fmaxf(xi, nx0);                       // clamp to n * x0_int
            float q  = floorf(xi * inv_x0);            // q in [0, 30]
            float r  = fmaf(-x0i, q, xi);              // xi - x0_int * q
            float z  = fmaf(r, r + bI, cI);            // r*(r + b_int) + c_int
            // z * 2^(n - q) computed exactly with v_ldexp_f32
            float e  = fmaxf(floorf(ldexpf(z, 30 - (int)q)), 0.0f);
            v[i][c]  = e;
            sum     += e;
        }
    }
    sum = wave32_sum(sum);

    // ---- Pass 3: log2-rounded power-of-two quantization, single HBM write ----
#pragma unroll
    for (int i = 0; i < 8; ++i) {
        v4f o;
#pragma unroll
        for (int c = 0; c < 4; ++c) {
            float e  = v[i][c];
            // sum/e via v_rcp_f32 (TRANS, co-executes with VALU) + round-half-even
            float so = rintf(sum * __builtin_amdgcn_rcpf(e));
            int   b  = __float_as_int(so);
            // floor(log2(so)) from exponent field; the reference's
            // (so - 2^big) >= 2^(big-1) test is exactly "mantissa >= 1.5" = bit 22.
            int rounds = (((b >> 23) & 0xff) - 127) + ((b >> 22) & 1);
            // e == 0 -> softmax_out = inf -> masked to 0 (matches reference)
            o[c] = (e > 0.0f && rounds < 256) ? ldexpf(1.0f, -rounds) : 0.0f;
        }
        __builtin_nontemporal_store(o, (v4f*)(op + i * 128));
    }
}

extern "C" void kernel_launch(void* const* d_in, const int* in_sizes, int n_in,
                              void* d_out, int out_size, void* d_ws, size_t ws_size,
                              hipStream_t stream) {
    const float* x     = (const float*)d_in[0];
    const float* scale = (const float*)d_in[1];
    float*       out   = (float*)d_out;

    const int total  = in_sizes[0];        // 8*12*1024*1024
    const int rows   = total >> 10;        // softmax rows of 1024
    const int blocks = (rows + 7) / 8;     // 8 rows (8 waves) per block

    hipLaunchKernelGGL(qintsoftmax_kernel, dim3(blocks), dim3(256), 0, stream,
                       x, scale, out, rows);
}